// QRNNLayer_19241453486609
// MI455X (gfx1250) — compile-verified
//
#include <hip/hip_runtime.h>
#include <hip/hip_bf16.h>

// Problem constants (from reference)
#define BQ   32            // batch
#define INC  1024          // input channels
#define HC   1024          // hidden
#define TT   512           // time
#define KW   3             // conv kernel width
#define CH3  (3*HC)        // 3072 output channels
#define KR   (INC*KW)      // 3072 reduction length

typedef float v2f __attribute__((ext_vector_type(2)));
typedef float v8f __attribute__((ext_vector_type(8)));

// ---------------- Phase 1: conv-as-GEMM with V_WMMA_F32_16X16X4_F32 --------
#define BM 128
#define BN 128
#define BK 32
#define LDA_P 36           // padded row pitch (floats) for A tile: 16B-aligned rows
#define LDB_P 288          // pair-row pitch (floats): 2*BN + 32 pad -> hi-half lanes
                           // land in banks 32..63, conflict-free b64 reads

__global__ __launch_bounds__(256) void qrnn_conv_gemm(
    const float* __restrict__ x,     // [B, IN, T]
    const float* __restrict__ w,     // [3H, IN, K] == row-major [3H, KR]
    const float* __restrict__ bias,  // [3H]
    float* __restrict__ gates)       // [B, 3H, T]
{
    __shared__ float ldsA[BM * LDA_P];          // 128 x 32 (padded)
    __shared__ float ldsB2[(BK/2) * LDB_P];     // k-paired: [pair][n][k&1]

    const int tid  = threadIdx.x;
    const int lane = tid & 31;
    const int wid  = tid >> 5;           // 0..7
    const int wm   = wid & 3;            // wave row   (4 in M)
    const int wn   = wid >> 2;           // wave col   (2 in N)
    const int hi   = lane >> 4;          // lane half
    const int ln   = lane & 15;

    // grid: b * 96 + mt*4 + nt   (24 M-tiles x 4 N-tiles per batch)
    const int blk = blockIdx.x;
    const int b   = blk / 96;
    const int mt  = (blk % 96) >> 2;
    const int nt  = blk & 3;
    const int m0  = mt * BM;
    const int t0  = nt * BN;

    const float* xb = x + (size_t)b * INC * TT;

    v8f acc[2][4];
#pragma unroll
    for (int s = 0; s < 2; ++s)
#pragma unroll
        for (int u = 0; u < 4; ++u)
            acc[s][u] = (v8f){0.f,0.f,0.f,0.f,0.f,0.f,0.f,0.f};

    for (int kb = 0; kb < KR; kb += BK) {
        // --- stage A tile: 128 rows x 32 k, float4 loads (4 per thread)
#pragma unroll
        for (int j = 0; j < 4; ++j) {
            int idx = tid + j * 256;         // 1024 float4 units
            int row = idx >> 3;
            int c4  = (idx & 7) << 2;
            float4 v = *(const float4*)(w + (size_t)(m0 + row) * KR + kb + c4);
            *(float4*)(&ldsA[row * LDA_P + c4]) = v;
        }
        // --- stage B tile (im2col), k-paired so fragments are single b64 loads
#pragma unroll
        for (int j = 0; j < 8; ++j) {
            int idx = tid + j * 256;         // 2048 float2 units
            int p   = idx >> 7;              // k-pair 0..15
            int col = idx & 127;             // local t
            int kr0 = kb + 2 * p;
            int ic0 = kr0 / KW, kk0 = kr0 - ic0 * KW;
            int kr1 = kr0 + 1;
            int ic1 = kr1 / KW, kk1 = kr1 - ic1 * KW;
            int tg0 = t0 + col + kk0 - (KW - 1);     // causal left shift
            int tg1 = t0 + col + kk1 - (KW - 1);
            float2 st;
            st.x = (tg0 >= 0) ? xb[(size_t)ic0 * TT + tg0] : 0.0f;
            st.y = (tg1 >= 0) ? xb[(size_t)ic1 * TT + tg1] : 0.0f;
            *(float2*)(&ldsB2[p * LDB_P + col * 2]) = st;
        }
        // --- prefetch next K-stage (global_prefetch_b8; L2-resident after pass 1)
        if (kb + BK < KR) {
            {
                int row = tid >> 1;
                int c4  = (tid & 1) << 4;
                __builtin_prefetch(w + (size_t)(m0 + row) * KR + (kb + BK) + c4, 0, 3);
            }
            {
                int p   = tid >> 4;          // spread over 16 pair-rows
                int col = (tid & 15) << 3;
                int kr0 = kb + BK + 2 * p;
                int ic0 = kr0 / KW;
                __builtin_prefetch(xb + (size_t)ic0 * TT + t0 + col, 0, 3);
            }
        }
        __syncthreads();

        // --- 8 k-steps of 4, 8 WMMAs each
#pragma unroll
        for (int k0 = 0; k0 < BK; k0 += 4) {
            v2f afr[2];
#pragma unroll
            for (int s = 0; s < 2; ++s) {
                int mrow = wm * 32 + s * 16 + ln;
                afr[s] = *(const v2f*)(&ldsA[mrow * LDA_P + k0 + 2 * hi]);
            }
#pragma unroll
            for (int u = 0; u < 4; ++u) {
                int ncol = wn * 64 + u * 16 + ln;
                v2f bfr = *(const v2f*)(&ldsB2[((k0 >> 1) + hi) * LDB_P + ncol * 2]);
#pragma unroll
                for (int s = 0; s < 2; ++s) {
                    acc[s][u] = __builtin_amdgcn_wmma_f32_16x16x4_f32(
                        false, afr[s], false, bfr, (short)0, acc[s][u],
                        false, false);
                }
            }
        }
        __syncthreads();
    }

    // --- epilogue: bias + activation (tanh for Z, sigmoid for F/O), store
#pragma unroll
    for (int s = 0; s < 2; ++s) {
#pragma unroll
        for (int u = 0; u < 4; ++u) {
            int tg = t0 + wn * 64 + u * 16 + ln;
#pragma unroll
            for (int r = 0; r < 8; ++r) {
                int ch = m0 + wm * 32 + s * 16 + r + 8 * hi;
                float v = acc[s][u][r] + bias[ch];
                v = (ch < HC) ? tanhf(v) : (1.0f / (1.0f + __expf(-v)));
                gates[((size_t)b * CH3 + ch) * TT + tg] = v;
            }
        }
    }
}

// ---------------- Phase 2: recurrent scan, LDS-transposed tiles ------------
#define P2H 128            // h lanes per block
#define P2T 32             // timesteps per LDS stage
#define PH  129            // LDS pitch over h (conflict-free both directions)

__global__ __launch_bounds__(128) void qrnn_scan(
    const float* __restrict__ gates,     // [B, 3H, T]
    const long long* __restrict__ ilen,  // [B] int64
    float* __restrict__ outC,            // [B, H, T]
    float* __restrict__ outH)            // [B, H, T]
{
    __shared__ float lz[P2T * PH];
    __shared__ float lf[P2T * PH];
    __shared__ float lo[P2T * PH];

    const int tid = threadIdx.x;
    const int b   = blockIdx.x >> 3;            // H/P2H = 8 blocks per batch
    const int h0  = (blockIdx.x & 7) * P2H;

    const long long len = ilen[b];

    const float* gz = gates + ((size_t)b * CH3 + 0 * HC + h0) * TT;
    const float* gf = gates + ((size_t)b * CH3 + 1 * HC + h0) * TT;
    const float* go = gates + ((size_t)b * CH3 + 2 * HC + h0) * TT;
    float* oc = outC + ((size_t)b * HC + h0) * TT;
    float* oh = outH + ((size_t)b * HC + h0) * TT;

    float c = 0.0f;

    for (int tc = 0; tc < TT; tc += P2T) {
        // coalesced float4 loads, transposed stores into LDS
#pragma unroll
        for (int j = 0; j < 8; ++j) {
            int idx4 = tid + j * 128;        // 1024 float4 = 128h x 8
            int hrow = idx4 >> 3;
            int t4   = (idx4 & 7) << 2;
            float4 vz = *(const float4*)(gz + (size_t)hrow * TT + tc + t4);
            float4 vf = *(const float4*)(gf + (size_t)hrow * TT + tc + t4);
            float4 vo = *(const float4*)(go + (size_t)hrow * TT + tc + t4);
            lz[(t4+0)*PH + hrow] = vz.x; lz[(t4+1)*PH + hrow] = vz.y;
            lz[(t4+2)*PH + hrow] = vz.z; lz[(t4+3)*PH + hrow] = vz.w;
            lf[(t4+0)*PH + hrow] = vf.x; lf[(t4+1)*PH + hrow] = vf.y;
            lf[(t4+2)*PH + hrow] = vf.z; lf[(t4+3)*PH + hrow] = vf.w;
            lo[(t4+0)*PH + hrow] = vo.x; lo[(t4+1)*PH + hrow] = vo.y;
            lo[(t4+2)*PH + hrow] = vo.z; lo[(t4+3)*PH + hrow] = vo.w;
        }
        __syncthreads();

        // recurrence: each thread owns one h lane, stride-1 LDS reads
#pragma unroll
        for (int tt = 0; tt < P2T; ++tt) {
            float z  = lz[tt * PH + tid];
            float f  = lf[tt * PH + tid];
            float o_ = lo[tt * PH + tid];
            c = c * f + z * (1.0f - f);          // carry stays unmasked
            float hval = c * o_;
            float m = ((long long)(tc + tt) < len) ? 1.0f : 0.0f;
            lz[tt * PH + tid] = c * m;           // masked C
            lf[tt * PH + tid] = hval * m;        // masked H
        }
        __syncthreads();

        // transposed gather from LDS, coalesced float4 stores
#pragma unroll
        for (int j = 0; j < 8; ++j) {
            int idx4 = tid + j * 128;
            int hrow = idx4 >> 3;
            int t4   = (idx4 & 7) << 2;
            float4 vc, vh;
            vc.x = lz[(t4+0)*PH + hrow]; vc.y = lz[(t4+1)*PH + hrow];
            vc.z = lz[(t4+2)*PH + hrow]; vc.w = lz[(t4+3)*PH + hrow];
            vh.x = lf[(t4+0)*PH + hrow]; vh.y = lf[(t4+1)*PH + hrow];
            vh.z = lf[(t4+2)*PH + hrow]; vh.w = lf[(t4+3)*PH + hrow];
            *(float4*)(oc + (size_t)hrow * TT + tc + t4) = vc;
            *(float4*)(oh + (size_t)hrow * TT + tc + t4) = vh;
        }
        __syncthreads();
    }
}

// ---------------------------------------------------------------------------
extern "C" void kernel_launch(void* const* d_in, const int* in_sizes, int n_in,
                              void* d_out, int out_size, void* d_ws, size_t ws_size,
                              hipStream_t stream) {
    const float*     x    = (const float*)d_in[0];       // [B, IN, T]
    const long long* ilen = (const long long*)d_in[1];   // [B] int64
    const float*     w    = (const float*)d_in[2];       // [3H, IN, K]
    const float*     bias = (const float*)d_in[3];       // [3H]

    float* gates = (float*)d_ws;                          // [B, 3H, T] scratch
    float* outC  = (float*)d_out;                         // [B, H, T]
    float* outH  = outC + (size_t)BQ * HC * TT;           // [B, H, T]

    // Phase 1: batched GEMM via fp32 WMMA (24 M-tiles x 4 N-tiles x 32 batches)
    qrnn_conv_gemm<<<BQ * 96, 256, 0, stream>>>(x, w, bias, gates);
    // Phase 2: bandwidth-bound recurrent scan
    qrnn_scan<<<BQ * 8, 128, 0, stream>>>(gates, ilen, outC, outH);
}